// YOLOLossV1_54966991454544
// MI455X (gfx1250) — compile-verified
//
#include <hip/hip_runtime.h>

// ---- problem constants (from reference) ----
#define N_IMGS   512
#define S_GRID   28
#define B_BOX    3
#define C_CLS    80
#define D_DIM    95                      // B*5 + C
#define N_CELL   (N_IMGS * S_GRID * S_GRID)   // 401408
#define TILE     128                     // cells per block
#define TPB      128                     // threads per block (4 waves, wave32)
#define TILE_F   (TILE * D_DIM)          // 12160 floats per tensor per tile
#define NT4      (TILE_F / 4)            // 3040 float4 chunks per tensor per tile
#define INV_S    (1.0f / 28.0f)
#define INV_N    (1.0f / 512.0f)
#define L_COORD  5.0f
#define L_NOOBJ  0.5f

// gfx1250 CPol: TH in [2:0]; TH=1 -> non-temporal (inputs are streamed once,
// no reuse -- don't let 305 MB of one-shot data churn the 192 MB L2).
#define CPOL_NT  1

typedef __attribute__((ext_vector_type(4))) float f32x4;
typedef __attribute__((ext_vector_type(4))) int   i32x4;

#if defined(__AMDGCN__) && __has_builtin(__builtin_amdgcn_global_load_async_to_lds_b128)
#define USE_ASYNC_LDS 1
typedef __attribute__((address_space(1))) i32x4 g_i32x4;   // global
typedef __attribute__((address_space(3))) i32x4 l_i32x4;   // LDS
#else
#define USE_ASYNC_LDS 0
#endif

__global__ void yolo_zero_out(float* out) { out[0] = 0.0f; }

__global__ __launch_bounds__(TPB) void yolo_loss_kernel(
    const float* __restrict__ pred,
    const float* __restrict__ targ,
    float* __restrict__ out)
{
    extern __shared__ float smem[];
    float* sP = smem;             // TILE_F floats
    float* sT = smem + TILE_F;    // TILE_F floats
    __shared__ float waveSums[TPB / 32];

    const int tid = threadIdx.x;
    const long long fBase = (long long)blockIdx.x * (long long)TILE_F;

    // ---------------- stage tile (coalesced b128, ASYNCcnt-tracked) --------
#if USE_ASYNC_LDS
    {
        g_i32x4* gP = (g_i32x4*)(pred + fBase);
        g_i32x4* gT = (g_i32x4*)(targ + fBase);
        l_i32x4* lP = (l_i32x4*)sP;
        l_i32x4* lT = (l_i32x4*)sT;
        for (int i = tid; i < NT4; i += TPB) {
            __builtin_amdgcn_global_load_async_to_lds_b128(gP + i, lP + i, 0, CPOL_NT);
            __builtin_amdgcn_global_load_async_to_lds_b128(gT + i, lT + i, 0, CPOL_NT);
        }
#if __has_builtin(__builtin_amdgcn_s_wait_asynccnt)
        __builtin_amdgcn_s_wait_asynccnt(0);
#else
        asm volatile("s_wait_asynccnt 0" ::: "memory");
#endif
    }
#else
    for (int i = tid; i < NT4; i += TPB) {
        *(f32x4*)(sP + 4 * i) = *(const f32x4*)(pred + fBase + 4 * i);
        *(f32x4*)(sT + 4 * i) = *(const f32x4*)(targ + fBase + 4 * i);
    }
#endif
    __syncthreads();

    // ---------------- per-cell loss ----------------
    const float* p  = sP + tid * D_DIM;
    const float* tg = sT + tid * D_DIM;

    float acc = 0.0f;

    const float tc0 = tg[0];
    const float cp0 = p[0], cp1 = p[1], cp2 = p[2];

    // no-object confidence loss (noobj mask: tconf0 == 0 -> conf_t == 0)
    if (tc0 == 0.0f) {
        float d0 = cp0 - tg[0];
        float d1 = cp1 - tg[1];
        float d2 = cp2 - tg[2];
        acc += L_NOOBJ * (d0 * d0 + d1 * d1 + d2 * d2);
    }

    if (tc0 > 0.0f) {
        // ground-truth box (box_t[...,0,:])
        const float tx = tg[3], ty = tg[4], tw = tg[5], th = tg[6];
        const float t1x = tx * INV_S - tw * 0.5f;
        const float t1y = ty * INV_S - th * 0.5f;
        const float t2x = tx * INV_S + tw * 0.5f;
        const float t2y = ty * INV_S + th * 0.5f;
        const float areaT = tw * th;

        float iou[B_BOX];
        float bx[B_BOX], by[B_BOX], bw[B_BOX], bh[B_BOX];
#pragma unroll
        for (int b = 0; b < B_BOX; ++b) {
            const float px = p[3 + 4 * b];
            const float py = p[4 + 4 * b];
            const float pw = p[5 + 4 * b];
            const float ph = p[6 + 4 * b];
            bx[b] = px; by[b] = py; bw[b] = pw; bh[b] = ph;
            const float p1x = px * INV_S - pw * 0.5f;
            const float p1y = py * INV_S - ph * 0.5f;
            const float p2x = px * INV_S + pw * 0.5f;
            const float p2y = py * INV_S + ph * 0.5f;
            const float ltx = fmaxf(p1x, t1x), lty = fmaxf(p1y, t1y);
            const float rbx = fminf(p2x, t2x), rby = fminf(p2y, t2y);
            const float wx = fmaxf(rbx - ltx, 0.0f);
            const float wy = fmaxf(rby - lty, 0.0f);
            const float inter = wx * wy;
            iou[b] = inter / (pw * ph + areaT - inter);
        }

        // argmax (first occurrence of max, matching jnp.argmax)
        int best = 0; float bi = iou[0];
        if (iou[1] > bi) { best = 1; bi = iou[1]; }
        if (iou[2] > bi) { best = 2; bi = iou[2]; }

        const float confp[B_BOX] = {cp0, cp1, cp2};

        // contain / not-contain confidence losses
        const float dc = confp[best] - bi;
        float sum_conf = dc * dc;               // contain_loss term
#pragma unroll
        for (int b = 0; b < B_BOX; ++b)
            if (b != best) sum_conf += confp[b] * confp[b];  // not_contain

        // localization loss (best box only; target box == gt for every b)
        const float dx = bx[best] - tx;
        const float dy = by[best] - ty;
        const float dw = sqrtf(bw[best]) - sqrtf(tw);
        const float dh = sqrtf(bh[best]) - sqrtf(th);
        const float loc = dx * dx + dy * dy + dw * dw + dh * dh;

        // class loss
        float cls = 0.0f;
#pragma unroll 8
        for (int c = 5 * B_BOX; c < D_DIM; ++c) {
            const float d = p[c] - tg[c];
            cls += d * d;
        }

        acc += L_COORD * loc + sum_conf + cls;
    }

    acc *= INV_N;

    // ---------------- reduction: wave32 shuffle -> LDS -> global atomic ----
#pragma unroll
    for (int off = 16; off > 0; off >>= 1)
        acc += __shfl_xor(acc, off, 32);

    if ((tid & 31) == 0) waveSums[tid >> 5] = acc;
    __syncthreads();

    if (tid == 0) {
        float s = 0.0f;
#pragma unroll
        for (int w = 0; w < TPB / 32; ++w) s += waveSums[w];
        __hip_atomic_fetch_add(out, s, __ATOMIC_RELAXED, __HIP_MEMORY_SCOPE_AGENT);
    }
}

extern "C" void kernel_launch(void* const* d_in, const int* in_sizes, int n_in,
                              void* d_out, int out_size, void* d_ws, size_t ws_size,
                              hipStream_t stream)
{
    const float* pred = (const float*)d_in[0];   // (N,S,S,D) f32
    const float* targ = (const float*)d_in[1];   // (N,S,S,D) f32
    float* out = (float*)d_out;                  // scalar f32

    yolo_zero_out<<<1, 1, 0, stream>>>(out);

    const int nTiles = N_CELL / TILE;            // 3136, exact
    const size_t smemBytes = (size_t)2 * TILE_F * sizeof(float);  // 97,280 B
    yolo_loss_kernel<<<nTiles, TPB, smemBytes, stream>>>(pred, targ, out);
}